// Repr1Classifier_7765300871371
// MI455X (gfx1250) — compile-verified
//
#include <hip/hip_runtime.h>
#include <hip/hip_bf16.h>

typedef __attribute__((ext_vector_type(2))) float v2f;
typedef __attribute__((ext_vector_type(8))) float v8f;

#define HDIM 64

// Fire-and-forget f32 atomic add at device scope: single CDNA5 VMEM RMW in L2
// (avoids HIP's CAS-loop fallback for float atomicAdd). Non-returning form is
// tracked by STOREcnt; S_ENDPGM performs an implicit wait-idle.
__device__ __forceinline__ void gadd_f32(float* p, float v) {
    asm volatile("global_atomic_add_f32 %0, %1, off scope:SCOPE_DEV"
                 :
                 : "v"(p), "v"(v)
                 : "memory");
}

// ---------------------------------------------------------------------------
// h0[n, 0:30] = x[n, :]; h0[n, 30:62] = emb_port[port[n]]; h0[n,62:64] = emb_flags[flag[n]]
__global__ __launch_bounds__(256) void k_embed(
        const float* __restrict__ x,
        const int* __restrict__ ports,
        const int* __restrict__ flags,
        const float* __restrict__ emb_port,
        const float* __restrict__ emb_flags,
        float* __restrict__ h, int N) {
    int i = blockIdx.x * blockDim.x + threadIdx.x;
    if (i >= N * HDIM) return;
    int n = i >> 6;
    int f = i & 63;
    float v;
    if (f < 30)      v = x[(size_t)n * 30 + f];
    else if (f < 62) v = emb_port[(size_t)ports[n] * 32 + (f - 30)];
    else             v = emb_flags[(size_t)flags[n] * 2 + (f - 62)];
    h[i] = v;
}

// ---------------------------------------------------------------------------
// One wave (32 lanes) per edge; lane t moves features 2t, 2t+1.
// agg[dst] += h[src]   via global_atomic_add_f32 (device scope, RMW in L2)
__global__ __launch_bounds__(256) void k_scatter(
        const float* __restrict__ h,
        const int* __restrict__ src,
        const int* __restrict__ dst,
        float* __restrict__ agg, int E) {
    int wave = (blockIdx.x * blockDim.x + threadIdx.x) >> 5;
    int lane = threadIdx.x & 31;
    if (wave >= E) return;
    int s = src[wave];
    int d = dst[wave];
    const float* hp = h + (size_t)s * HDIM + lane * 2;
    float2 v = *(const float2*)hp;          // one global_load_b64 per lane
    float* ap = agg + (size_t)d * HDIM + lane * 2;
    gadd_f32(ap + 0, v.x);
    gadd_f32(ap + 1, v.y);
}

// ---------------------------------------------------------------------------
// One wave computes a 16x16 tile of:  relu(agg @ Wrel^T + brel + h @ Wroot^T)
// Using V_WMMA_F32_16X16X4_F32, K=64 in 16 steps of 4 (two chained MACs/step).
// ISA layouts (wave32): A 16x4 f32: lanes 0-15 hold M=0..15, VGPR j = K=j
//   (half g adds 2*g to K). B 4x16: VGPR j, lanes[0:16)=row K=j+2g, col=lane&15.
// B[k][o] = W[o][k] with W row-major -> contiguous float2 per lane.
__global__ __launch_bounds__(256) void k_graphconv(
        const float* __restrict__ agg,
        const float* __restrict__ h,
        const float* __restrict__ Wrel,
        const float* __restrict__ brel,
        const float* __restrict__ Wroot,
        float* __restrict__ hout, int N) {
    int tile = (blockIdx.x * blockDim.x + threadIdx.x) >> 5;
    int lane = threadIdx.x & 31;
    int ntiles = (N >> 4) * 4;
    if (tile >= ntiles) return;            // whole-wave exit: EXEC stays all-1
    int tm = tile >> 2;
    int tn = tile & 3;
    int row0 = tm * 16;
    int col0 = tn * 16;
    int m = lane & 15;
    int g = lane >> 4;

    const float* arow = agg   + (size_t)(row0 + m) * HDIM + g * 2;
    const float* hrow = h     + (size_t)(row0 + m) * HDIM + g * 2;
    const float* wrl  = Wrel  + (size_t)(col0 + m) * HDIM + g * 2;
    const float* wrt  = Wroot + (size_t)(col0 + m) * HDIM + g * 2;

    v8f c = {};
#pragma unroll
    for (int kk = 0; kk < 16; ++kk) {
        int kb = kk * 4;
        v2f a0 = *(const v2f*)(arow + kb);
        v2f b0 = *(const v2f*)(wrl + kb);
        v2f a1 = *(const v2f*)(hrow + kb);
        v2f b1 = *(const v2f*)(wrt + kb);
        c = __builtin_amdgcn_wmma_f32_16x16x4_f32(false, a0, false, b0,
                                                  (short)0, c, false, false);
        c = __builtin_amdgcn_wmma_f32_16x16x4_f32(false, a1, false, b1,
                                                  (short)0, c, false, false);
    }
    float bias = brel[col0 + m];
#pragma unroll
    for (int r = 0; r < 8; ++r) {
        float v = c[r] + bias;
        v = v > 0.0f ? v : 0.0f;
        // C/D layout: VGPR r, lanes 0-15 -> M=r; lanes 16-31 -> M=r+8; N=lane&15
        hout[(size_t)(row0 + r + g * 8) * HDIM + col0 + m] = v;
    }
}

// ---------------------------------------------------------------------------
// segment_max over batch. h >= 0 after relu, so IEEE bits compare as ints.
__global__ __launch_bounds__(256) void k_pool(
        const float* __restrict__ h,
        const int* __restrict__ batch,
        float* __restrict__ pool, int N) {
    int i = blockIdx.x * blockDim.x + threadIdx.x;
    if (i >= N * HDIM) return;
    int n = i >> 6;
    int f = i & 63;
    int b = batch[n];
    atomicMax((int*)&pool[(size_t)b * HDIM + f], __float_as_int(h[i]));
}

// ---------------------------------------------------------------------------
// out = relu(in @ W^T + b), in: [rows, 64], W: [64, 64] row-major (out, in)
__global__ __launch_bounds__(256) void k_fc(
        const float* __restrict__ in,
        const float* __restrict__ W,
        const float* __restrict__ b,
        float* __restrict__ out, int rows) {
    int tile = (blockIdx.x * blockDim.x + threadIdx.x) >> 5;
    int lane = threadIdx.x & 31;
    int ntiles = (rows >> 4) * 4;
    if (tile >= ntiles) return;
    int tm = tile >> 2;
    int tn = tile & 3;
    int row0 = tm * 16;
    int col0 = tn * 16;
    int m = lane & 15;
    int g = lane >> 4;

    const float* arow = in + (size_t)(row0 + m) * HDIM + g * 2;
    const float* wr   = W  + (size_t)(col0 + m) * HDIM + g * 2;

    v8f c = {};
#pragma unroll
    for (int kk = 0; kk < 16; ++kk) {
        int kb = kk * 4;
        v2f a0 = *(const v2f*)(arow + kb);
        v2f b0 = *(const v2f*)(wr + kb);
        c = __builtin_amdgcn_wmma_f32_16x16x4_f32(false, a0, false, b0,
                                                  (short)0, c, false, false);
    }
    float bias = b[col0 + m];
#pragma unroll
    for (int r = 0; r < 8; ++r) {
        float v = c[r] + bias;
        v = v > 0.0f ? v : 0.0f;
        out[(size_t)(row0 + r + g * 8) * HDIM + col0 + m] = v;
    }
}

// ---------------------------------------------------------------------------
// Final head: out[r, c] = g2[r,:] . fc3_W[c,:] + fc3_b[c],  128x10, scalar.
__global__ __launch_bounds__(256) void k_fc3(
        const float* __restrict__ in,
        const float* __restrict__ W,
        const float* __restrict__ b,
        float* __restrict__ out, int rows, int C) {
    int i = blockIdx.x * blockDim.x + threadIdx.x;
    if (i >= rows * C) return;
    int r = i / C;
    int c = i % C;
    float acc = b[c];
#pragma unroll
    for (int k = 0; k < HDIM; ++k)
        acc += in[(size_t)r * HDIM + k] * W[(size_t)c * HDIM + k];
    out[i] = acc;
}

// ---------------------------------------------------------------------------
extern "C" void kernel_launch(void* const* d_in, const int* in_sizes, int n_in,
                              void* d_out, int out_size, void* d_ws, size_t ws_size,
                              hipStream_t stream) {
    const float* x        = (const float*)d_in[0];
    const int*   ports    = (const int*)d_in[1];
    const int*   flags    = (const int*)d_in[2];
    const int*   eidx     = (const int*)d_in[3];
    const int*   batch    = (const int*)d_in[4];
    const float* emb_port = (const float*)d_in[5];
    const float* emb_flag = (const float*)d_in[6];
    const float* Wrel     = (const float*)d_in[7];
    const float* brel     = (const float*)d_in[8];
    const float* Wroot    = (const float*)d_in[9];
    const float* fc1W     = (const float*)d_in[10];
    const float* fc1b     = (const float*)d_in[11];
    const float* fc2W     = (const float*)d_in[12];
    const float* fc2b     = (const float*)d_in[13];
    const float* fc3W     = (const float*)d_in[14];
    const float* fc3b     = (const float*)d_in[15];

    const int N = in_sizes[1];          // dst_ports: N entries
    const int E = in_sizes[3] / 2;      // edge_index: [2, E]
    const int B = 128;
    const int L = 3;

    // workspace carve-up (floats): hA, hB, agg: N*64 each; pool/g1/g2: B*64
    float* hA   = (float*)d_ws;
    float* hB   = hA  + (size_t)N * HDIM;
    float* agg  = hB  + (size_t)N * HDIM;
    float* pool = agg + (size_t)N * HDIM;
    float* g1   = pool + (size_t)B * HDIM;
    float* g2   = g1   + (size_t)B * HDIM;

    // 1) embedding + concat
    {
        int total = N * HDIM;
        k_embed<<<(total + 255) / 256, 256, 0, stream>>>(
            x, ports, flags, emb_port, emb_flag, hA, N);
    }

    const int* src = eidx;
    const int* dst = eidx + E;
    float* cur = hA;
    float* nxt = hB;

    // 2) three GraphConv layers
    for (int l = 0; l < L; ++l) {
        hipMemsetAsync(agg, 0, (size_t)N * HDIM * sizeof(float), stream);
        {
            long long threads = (long long)E * 32;
            k_scatter<<<(int)((threads + 255) / 256), 256, 0, stream>>>(
                cur, src, dst, agg, E);
        }
        {
            int ntiles = (N >> 4) * 4;                 // N divisible by 16
            int threads = ntiles * 32;
            k_graphconv<<<(threads + 255) / 256, 256, 0, stream>>>(
                agg, cur, Wrel + (size_t)l * HDIM * HDIM, brel + (size_t)l * HDIM,
                Wroot + (size_t)l * HDIM * HDIM, nxt, N);
        }
        float* t = cur; cur = nxt; nxt = t;
    }

    // 3) global max pool (relu outputs >= 0, init 0)
    hipMemsetAsync(pool, 0, (size_t)B * HDIM * sizeof(float), stream);
    {
        int total = N * HDIM;
        k_pool<<<(total + 255) / 256, 256, 0, stream>>>(cur, batch, pool, N);
    }

    // 4) MLP head
    {
        int ntiles = (B >> 4) * 4;                     // 32 tiles
        int threads = ntiles * 32;
        k_fc<<<(threads + 255) / 256, 256, 0, stream>>>(pool, fc1W, fc1b, g1, B);
        k_fc<<<(threads + 255) / 256, 256, 0, stream>>>(g1, fc2W, fc2b, g2, B);
    }
    {
        int total = B * 10;
        k_fc3<<<(total + 255) / 256, 256, 0, stream>>>(g2, fc3W, fc3b,
                                                       (float*)d_out, B, 10);
    }
}